// ShadingCompositor_88158498718276
// MI455X (gfx1250) — compile-verified
//
#include <hip/hip_runtime.h>
#include <hip/hip_bf16.h>
#include <math.h>

// Problem geometry (fixed by the reference): P=100000, N=8, H=W=256, K=8
#define TILE        256
#define KNN         8
#define NPIX        (8 * 256 * 256)
#define NTILES      (NPIX / TILE)      // 2048
#define GRID_BLOCKS 512                // persistent blocks -> 4 tiles each, double-buffered async pipeline

struct F3 { float x, y, z; };          // packed 12-byte float3 (HIP float3 is 16B ext-vector)

// ---------------- CDNA5 async global->LDS staging (ASYNCcnt path) ----------------
#if defined(__gfx1250__)
#define USE_ASYNC 1

typedef __attribute__((ext_vector_type(4))) int v4i;
typedef __attribute__((address_space(1))) v4i*  gv4i_p;   // global (HIP __device__) AS
typedef __attribute__((address_space(3))) v4i*  lv4i_p;   // LDS (HIP __shared__) AS
typedef __attribute__((address_space(3))) void  LDSV;

__device__ __forceinline__ void async_cp16(const void* g, void* l) {
#if __has_builtin(__builtin_amdgcn_global_load_async_to_lds_b128)
  v4i* gp = (v4i*)g;   // C-style cast drops const + retypes
  v4i* lp = (v4i*)l;
  __builtin_amdgcn_global_load_async_to_lds_b128((gv4i_p)gp, (lv4i_p)lp, 0, 0);
#else
  unsigned int       loff = (unsigned int)(unsigned long long)(LDSV*)l; // raw LDS byte offset
  unsigned long long ga   = (unsigned long long)g;
  asm volatile("global_load_async_to_lds_b128 %0, %1, off"
               :: "v"(loff), "v"(ga) : "memory");
#endif
}

#if __has_builtin(__builtin_amdgcn_s_wait_asynccnt)
#define WAIT_ASYNC(n) __builtin_amdgcn_s_wait_asynccnt(n)
#else
#define WAIT_ASYNC(n) asm volatile("s_wait_asynccnt " #n ::: "memory")
#endif
#endif // __gfx1250__

// -------------------------------- kernel --------------------------------
extern "C" __global__ void __launch_bounds__(TILE)
shading_compositor_kernel(const int*   __restrict__ gIdx,
                          const float* __restrict__ gDst,
                          const float* __restrict__ pts,
                          const float* __restrict__ nrm,
                          const float* __restrict__ lLoc,
                          const float* __restrict__ lAmb,
                          const float* __restrict__ lDif,
                          const float* __restrict__ lSpc,
                          const float* __restrict__ mAmb,
                          const float* __restrict__ mDif,
                          const float* __restrict__ mSpc,
                          const float* __restrict__ cam,
                          const int*   __restrict__ shinP,
                          float*       __restrict__ out)
{
  const int tid = threadIdx.x;

  // Uniform shading constants (uniform addresses -> scalar loads, L2/K$ hits)
  const float llx = lLoc[0], lly = lLoc[1], llz = lLoc[2];
  const float cpx = cam[0],  cpy = cam[1],  cpz = cam[2];
  const float amb0 = lAmb[0]*mAmb[0], amb1 = lAmb[1]*mAmb[1], amb2 = lAmb[2]*mAmb[2];
  const float kd0  = lDif[0]*mDif[0], kd1  = lDif[1]*mDif[1], kd2  = lDif[2]*mDif[2];
  const float ks0  = lSpc[0]*mSpc[0], ks1  = lSpc[1]*mSpc[1], ks2  = lSpc[2]*mSpc[2];
  const int   shin  = shinP[0];
  const bool  pow64 = (shin == 64);    // wave-uniform; reference uses shininess = 64

#ifdef USE_ASYNC
  __shared__ __align__(16) int   sIdx[2][TILE * KNN];   // 2 x 8 KB
  __shared__ __align__(16) float sDst[2][TILE * KNN];   // 2 x 8 KB
  {
    // Prologue: stage first tile into buffer 0 (4 async b128 per thread)
    const size_t base = (size_t)blockIdx.x * (TILE * KNN) + (size_t)tid * KNN;
    async_cp16(gIdx + base,     &sIdx[0][tid * KNN]);
    async_cp16(gIdx + base + 4, &sIdx[0][tid * KNN + 4]);
    async_cp16(gDst + base,     &sDst[0][tid * KNN]);
    async_cp16(gDst + base + 4, &sDst[0][tid * KNN + 4]);
  }
#endif

  int buf = 0;
  for (int tile = blockIdx.x; tile < NTILES; tile += gridDim.x) {
    int   i8[KNN];
    float dv[KNN];

#ifdef USE_ASYNC
    const int nxt = tile + gridDim.x;
    if (nxt < NTILES) {
      // Issue next tile into the other buffer, then release current tile.
      // Async loads retire in order, so asynccnt<=4 == "first 4 (current tile) done".
      const size_t base = (size_t)nxt * (TILE * KNN) + (size_t)tid * KNN;
      const int nb = buf ^ 1;
      async_cp16(gIdx + base,     &sIdx[nb][tid * KNN]);
      async_cp16(gIdx + base + 4, &sIdx[nb][tid * KNN + 4]);
      async_cp16(gDst + base,     &sDst[nb][tid * KNN]);
      async_cp16(gDst + base + 4, &sDst[nb][tid * KNN + 4]);
      WAIT_ASYNC(4);
    } else {
      WAIT_ASYNC(0);
    }
    asm volatile("" ::: "memory");
    const int4   ia = *(const int4*)  &sIdx[buf][tid * KNN];
    const int4   ib = *(const int4*)  &sIdx[buf][tid * KNN + 4];
    const float4 da = *(const float4*)&sDst[buf][tid * KNN];
    const float4 db = *(const float4*)&sDst[buf][tid * KNN + 4];
#else
    const size_t base = (size_t)tile * (TILE * KNN) + (size_t)tid * KNN;
    const int4   ia = *(const int4*)  (gIdx + base);
    const int4   ib = *(const int4*)  (gIdx + base + 4);
    const float4 da = *(const float4*)(gDst + base);
    const float4 db = *(const float4*)(gDst + base + 4);
#endif
    i8[0] = ia.x; i8[1] = ia.y; i8[2] = ia.z; i8[3] = ia.w;
    i8[4] = ib.x; i8[5] = ib.y; i8[6] = ib.z; i8[7] = ib.w;
    dv[0] = da.x; dv[1] = da.y; dv[2] = da.z; dv[3] = da.w;
    dv[4] = db.x; dv[5] = db.y; dv[6] = db.z; dv[7] = db.w;

    // L1-normalized weights over K: wn = dm / max(sum(|dm|), 1e-12), dm = (d != -1) ? d : 0
    float denom = 0.0f;
#pragma unroll
    for (int k = 0; k < KNN; ++k) {
      float d = dv[k];
      d = (d != -1.0f) ? d : 0.0f;
      dv[k] = d;
      denom += fabsf(d);
    }
    const float winv = 1.0f / fmaxf(denom, 1e-12f);

    float a0 = 0.0f, a1 = 0.0f, a2 = 0.0f;
#pragma unroll
    for (int k = 0; k < KNN; ++k) {
      int i = i8[k];
      i = (i < 0) ? 0 : i;                       // masked gather: idx<0 reads point 0
      const F3 p = ((const F3*)pts)[i];          // b96 gathers, L2-resident tables
      const F3 n = ((const F3*)nrm)[i];

      // dir to light, normalize with clip(norm, 1e-6)
      float lx = llx - p.x, ly = lly - p.y, lz = llz - p.z;
      const float li = 1.0f / fmaxf(sqrtf(lx*lx + ly*ly + lz*lz), 1e-6f);
      lx *= li; ly *= li; lz *= li;

      const float cosv = n.x*lx + n.y*ly + n.z*lz;
      const float rcos = fmaxf(cosv, 0.0f);

      // view dir
      float vx = cpx - p.x, vy = cpy - p.y, vz = cpz - p.z;
      const float vi = 1.0f / fmaxf(sqrtf(vx*vx + vy*vy + vz*vz), 1e-6f);
      vx *= vi; vy *= vi; vz *= vi;

      // reflect = 2*cos*n - l ; alpha = relu(dot(view, reflect)) * (cos > 0)
      const float t2 = 2.0f * cosv;
      const float rx = t2*n.x - lx, ry = t2*n.y - ly, rz = t2*n.z - lz;
      float alpha = fmaxf(vx*rx + vy*ry + vz*rz, 0.0f);
      alpha = (cosv > 0.0f) ? alpha : 0.0f;

      // alpha ** shininess: exact integer power. Fast straight-line path for the
      // uniform shin==64 case (6 squarings); branchy generic loop only as fallback.
      float s;
      if (pow64) {
        float b2  = alpha * alpha;
        float b4  = b2  * b2;
        float b8  = b4  * b4;
        float b16 = b8  * b8;
        float b32 = b16 * b16;
        s         = b32 * b32;         // alpha^64
      } else {
        s = 1.0f;
        float b = alpha;
        int e = shin;
        while (e) { if (e & 1) s *= b; b *= b; e >>= 1; }
      }

      const float w = dv[k] * winv;
      a0 += w * (amb0 + kd0 * rcos + ks0 * s);
      a1 += w * (amb1 + kd1 * rcos + ks1 * s);
      a2 += w * (amb2 + kd2 * rcos + ks2 * s);
    }

    // (N,H,W,3): 12 contiguous bytes per lane, fully coalesced b96 stores
    ((F3*)out)[tile * TILE + tid] = F3{a0, a1, a2};
    buf ^= 1;
  }
}

// -------------------------------- launch --------------------------------
extern "C" void kernel_launch(void* const* d_in, const int* in_sizes, int n_in,
                              void* d_out, int out_size, void* d_ws, size_t ws_size,
                              hipStream_t stream) {
  const int*   idx  = (const int*)  d_in[0];
  const float* dst  = (const float*)d_in[1];
  const float* pts  = (const float*)d_in[2];
  const float* nrm  = (const float*)d_in[3];
  const float* ll   = (const float*)d_in[4];
  const float* la   = (const float*)d_in[5];
  const float* ld   = (const float*)d_in[6];
  const float* ls   = (const float*)d_in[7];
  const float* ma   = (const float*)d_in[8];
  const float* md   = (const float*)d_in[9];
  const float* ms   = (const float*)d_in[10];
  const float* cam  = (const float*)d_in[11];
  const int*   shin = (const int*)  d_in[12];
  float* out = (float*)d_out;

  const int grid = (NTILES < GRID_BLOCKS) ? NTILES : GRID_BLOCKS;
  hipLaunchKernelGGL(shading_compositor_kernel, dim3(grid), dim3(TILE), 0, stream,
                     idx, dst, pts, nrm, ll, la, ld, ls, ma, md, ms, cam, shin, out);
}